// GAT_28123445854591
// MI455X (gfx1250) — compile-verified
//
#include <hip/hip_runtime.h>
#include <cstddef>

// ---------------- problem constants (match reference) ----------------
#define NN 100000      // nodes
#define EE 1600000     // directed edges (before self loops)
#define ETOT 1700000   // EE + NN (with self loops)
#define DD 128         // hidden / input feature dim
#define GG 64          // graphs
#define CC 10          // classes
#define NLAYER 4
#define STAT_BLOCKS 625
#define STAT_CHUNK 160   // 625*160 == 100000 exactly

typedef __attribute__((ext_vector_type(2))) float v2f;
typedef __attribute__((ext_vector_type(8))) float v8f;

// order-preserving float<->uint encoding so we can use integer atomicMax
__device__ __forceinline__ unsigned f2o(float f) {
  unsigned b = __float_as_uint(f);
  return (b & 0x80000000u) ? ~b : (b | 0x80000000u);
}
__device__ __forceinline__ float o2f(unsigned u) {
  unsigned b = (u & 0x80000000u) ? (u & 0x7fffffffu) : ~u;
  return __uint_as_float(b);
}

// ---------------- utility kernels ----------------
__global__ void k_zero_f32(float* __restrict__ p, int n) {
  int i = blockIdx.x * blockDim.x + threadIdx.x;
  int stride = gridDim.x * blockDim.x;
  for (; i < n; i += stride) p[i] = 0.f;
}

__global__ void k_zero2_128(float* __restrict__ a, float* __restrict__ b) {
  a[threadIdx.x] = 0.f;
  b[threadIdx.x] = 0.f;
}

__global__ void k_init_ms(unsigned* __restrict__ menc, float* __restrict__ s) {
  int i = blockIdx.x * blockDim.x + threadIdx.x;
  if (i < NN) { menc[i] = 0x007FFFFFu; /* f2o(-inf) */ s[i] = 0.f; }
}

// ---------------- fp32 WMMA GEMM: out[N,128] = A[N,128] @ W[128,128] --------
// block = 128 threads (4 waves). Block covers 32 rows; wave w handles
// rows (w>>1)*16 and column half (w&1)*64 (four 16x16 WMMA tiles).
// W is staged in LDS (64 KB). 3125 blocks * 32 rows == 100000 exactly.
__global__ __launch_bounds__(128) void k_gemm128_wmma(
    const float* __restrict__ A, const float* __restrict__ W,
    float* __restrict__ out) {
  __shared__ float sW[DD * DD];  // 64 KB of the 320 KB WGP LDS
  int tid = threadIdx.x;
  for (int i = tid; i < DD * DD; i += 128) sW[i] = W[i];
  __syncthreads();

  int wave = tid >> 5, lane = tid & 31;
  int m_base = blockIdx.x * 32 + (wave >> 1) * 16;
  int c_base = (wave & 1) * 64;
  int mrow = lane & 15;              // M index within tile (A) / N index (B,C)
  int khalf = (lane >> 4) << 1;      // second half-wave holds K+2,K+3

  const float* arow = A + (size_t)(m_base + mrow) * DD;
  v8f acc0 = {}, acc1 = {}, acc2 = {}, acc3 = {};

#pragma unroll 4
  for (int k0 = 0; k0 < DD; k0 += 4) {
    int kk = k0 + khalf;
    v2f a = *(const v2f*)(arow + kk);  // A[m, kk], A[m, kk+1]
    int col = c_base + mrow;
    v2f b0, b1, b2, b3;
    b0.x = sW[kk * DD + col];      b0.y = sW[(kk + 1) * DD + col];
    b1.x = sW[kk * DD + col + 16]; b1.y = sW[(kk + 1) * DD + col + 16];
    b2.x = sW[kk * DD + col + 32]; b2.y = sW[(kk + 1) * DD + col + 32];
    b3.x = sW[kk * DD + col + 48]; b3.y = sW[(kk + 1) * DD + col + 48];
    acc0 = __builtin_amdgcn_wmma_f32_16x16x4_f32(false, a, false, b0, (short)0, acc0, false, false);
    acc1 = __builtin_amdgcn_wmma_f32_16x16x4_f32(false, a, false, b1, (short)0, acc1, false, false);
    acc2 = __builtin_amdgcn_wmma_f32_16x16x4_f32(false, a, false, b2, (short)0, acc2, false, false);
    acc3 = __builtin_amdgcn_wmma_f32_16x16x4_f32(false, a, false, b3, (short)0, acc3, false, false);
  }

  // C/D layout: VGPR v -> M = v + 8*(lane>=16), N = lane%16
  int nlo = lane & 15;
  int moff = (lane >> 4) << 3;
#pragma unroll
  for (int v = 0; v < 8; v++) {
    float* orow = out + (size_t)(m_base + moff + v) * DD + c_base + nlo;
    orow[0]  = acc0[v];
    orow[16] = acc1[v];
    orow[32] = acc2[v];
    orow[48] = acc3[v];
  }
}

// ---------------- per-node attention scores: e_src = h@a_src, e_dst = h@a_dst
// one wave per row; 12500 blocks * 8 warps == 100000 exactly
__global__ __launch_bounds__(256) void k_row_scores(
    const float* __restrict__ h, const float* __restrict__ asrc,
    const float* __restrict__ adst, float* __restrict__ esrc,
    float* __restrict__ edst) {
  int warp = threadIdx.x >> 5, lane = threadIdx.x & 31;
  int row = blockIdx.x * 8 + warp;
  const float* hr = h + (size_t)row * DD;
  float s0 = 0.f, s1 = 0.f;
#pragma unroll
  for (int c = lane; c < DD; c += 32) {
    float v = hr[c];
    s0 = fmaf(v, asrc[c], s0);
    s1 = fmaf(v, adst[c], s1);
  }
#pragma unroll
  for (int off = 16; off > 0; off >>= 1) {
    s0 += __shfl_xor(s0, off, 32);
    s1 += __shfl_xor(s1, off, 32);
  }
  if (lane == 0) { esrc[row] = s0; edst[row] = s1; }
}

// ---------------- edge pipeline (segment softmax over dst) ----------------
__device__ __forceinline__ void edge_endpoints(const long long* __restrict__ ei,
                                               int e, int& s, int& d) {
  if (e < EE) { s = (int)ei[e]; d = (int)ei[EE + e]; }
  else        { s = e - EE;     d = e - EE; }        // self loop
}

__global__ void k_edge_max(const long long* __restrict__ ei,
                           const float* __restrict__ esrc,
                           const float* __restrict__ edst,
                           float* __restrict__ eval,
                           unsigned* __restrict__ menc) {
  int e = blockIdx.x * blockDim.x + threadIdx.x;
  if (e >= ETOT) return;
  int s, d; edge_endpoints(ei, e, s, d);
  float v = esrc[s] + edst[d];
  v = (v > 0.f) ? v : 0.2f * v;  // leaky_relu slope 0.2
  eval[e] = v;
  atomicMax(&menc[d], f2o(v));
}

__global__ void k_edge_exp(const long long* __restrict__ ei,
                           float* __restrict__ eval,
                           const unsigned* __restrict__ menc,
                           float* __restrict__ ssum) {
  int e = blockIdx.x * blockDim.x + threadIdx.x;
  if (e >= ETOT) return;
  int s, d; edge_endpoints(ei, e, s, d);
  float p = __expf(eval[e] - o2f(menc[d]));
  eval[e] = p;
  atomicAdd(&ssum[d], p);
}

// out[dst] += alpha * h[src]; block of 256 handles 2 edges, 128 feats each
__global__ __launch_bounds__(256) void k_aggregate(
    const long long* __restrict__ ei, const float* __restrict__ eval,
    const float* __restrict__ ssum, const float* __restrict__ hlin,
    float* __restrict__ agg) {
  int e = blockIdx.x * 2 + (threadIdx.x >> 7);
  int f = threadIdx.x & 127;
  if (e >= ETOT) return;
  int s, d; edge_endpoints(ei, e, s, d);
  float alpha = eval[e] / (ssum[d] + 1e-16f);
  atomicAdd(&agg[(size_t)d * DD + f], alpha * hlin[(size_t)s * DD + f]);
}

// ---------------- training-mode BatchNorm (+ReLU) with fused JK pooling -----
__global__ __launch_bounds__(128) void k_bn_stats(const float* __restrict__ h,
                                                  float* __restrict__ csum,
                                                  float* __restrict__ csq) {
  int j = threadIdx.x;
  int r0 = blockIdx.x * STAT_CHUNK;
  float s = 0.f, q = 0.f;
  for (int r = r0; r < r0 + STAT_CHUNK; r++) {
    float v = h[(size_t)r * DD + j];
    s += v;
    q = fmaf(v, v, q);
  }
  atomicAdd(&csum[j], s);
  atomicAdd(&csq[j], q);
}

__global__ void k_bn_finalize(const float* __restrict__ csum,
                              const float* __restrict__ csq,
                              float* __restrict__ mu, float* __restrict__ rstd) {
  int j = threadIdx.x;
  float m = csum[j] * (1.f / NN);
  float v = csq[j] * (1.f / NN) - m * m;
  v = v > 0.f ? v : 0.f;
  mu[j] = m;
  rstd[j] = rsqrtf(v + 1e-5f);
}

// note: conv bias cancels exactly through training-mode BN, so it is omitted.
// batch is sorted -> per-thread running accumulator, flush on graph change.
__global__ __launch_bounds__(128) void k_bn_apply_pool(
    const float* __restrict__ agg, const float* __restrict__ gamma,
    const float* __restrict__ beta, const float* __restrict__ mu,
    const float* __restrict__ rstd, float* __restrict__ hout,
    const long long* __restrict__ batch, float* __restrict__ pooled) {
  int j = threadIdx.x;
  int r0 = blockIdx.x * STAT_CHUNK;
  float ga = gamma[j], be = beta[j], m = mu[j], rs = rstd[j];
  int curg = -1;
  float acc = 0.f;
  for (int r = r0; r < r0 + STAT_CHUNK; r++) {
    float v = fmaf((agg[(size_t)r * DD + j] - m) * rs, ga, be);
    v = v > 0.f ? v : 0.f;  // relu
    hout[(size_t)r * DD + j] = v;
    int g = (int)batch[r];
    if (g != curg) {
      if (curg >= 0) atomicAdd(&pooled[curg * DD + j], acc);
      curg = g;
      acc = 0.f;
    }
    acc += v;
  }
  if (curg >= 0) atomicAdd(&pooled[curg * DD + j], acc);
}

__global__ __launch_bounds__(128) void k_pool_x(
    const float* __restrict__ x, const long long* __restrict__ batch,
    float* __restrict__ pooled) {
  int j = threadIdx.x;
  int r0 = blockIdx.x * STAT_CHUNK;
  int curg = -1;
  float acc = 0.f;
  for (int r = r0; r < r0 + STAT_CHUNK; r++) {
    float v = x[(size_t)r * DD + j];
    int g = (int)batch[r];
    if (g != curg) {
      if (curg >= 0) atomicAdd(&pooled[curg * DD + j], acc);
      curg = g;
      acc = 0.f;
    }
    acc += v;
  }
  if (curg >= 0) atomicAdd(&pooled[curg * DD + j], acc);
}

// ---------------- jumping-knowledge heads + log_softmax ----------------
__global__ __launch_bounds__(32) void k_jk_out(
    const float* __restrict__ pooled, const float* w0, const float* w1,
    const float* w2, const float* w3, const float* w4, const float* b0,
    const float* b1, const float* b2, const float* b3, const float* b4,
    float* __restrict__ out) {
  __shared__ float sl[CC];
  __shared__ float red[2];
  int g = blockIdx.x, c = threadIdx.x;
  const float* Wh[5] = {w0, w1, w2, w3, w4};
  const float* Bh[5] = {b0, b1, b2, b3, b4};
  if (c < CC) {
    float acc = 0.f;
    for (int h = 0; h < 5; h++) {
      const float* pg = pooled + (size_t)h * GG * DD + (size_t)g * DD;
      const float* W = Wh[h];
      float a = 0.f;
      for (int f = 0; f < DD; f++) a = fmaf(pg[f], W[f * CC + c], a);
      acc += a + Bh[h][c];
    }
    sl[c] = acc;
  }
  __syncthreads();
  if (c == 0) {
    float mx = sl[0];
    for (int i = 1; i < CC; i++) mx = fmaxf(mx, sl[i]);
    float ss = 0.f;
    for (int i = 0; i < CC; i++) ss += __expf(sl[i] - mx);
    red[0] = mx;
    red[1] = __logf(ss);
  }
  __syncthreads();
  if (c < CC) out[g * CC + c] = sl[c] - red[0] - red[1];
}

// ---------------- host orchestration ----------------
extern "C" void kernel_launch(void* const* d_in, const int* in_sizes, int n_in,
                              void* d_out, int out_size, void* d_ws,
                              size_t ws_size, hipStream_t stream) {
  (void)in_sizes; (void)n_in; (void)out_size; (void)ws_size;
  const float* x = (const float*)d_in[0];
  const float* Ws[4]   = {(const float*)d_in[1],  (const float*)d_in[2],
                          (const float*)d_in[3],  (const float*)d_in[4]};
  const float* asrc[4] = {(const float*)d_in[5],  (const float*)d_in[6],
                          (const float*)d_in[7],  (const float*)d_in[8]};
  const float* adst[4] = {(const float*)d_in[9],  (const float*)d_in[10],
                          (const float*)d_in[11], (const float*)d_in[12]};
  // d_in[13..16] = conv biases: cancel exactly through training-mode BN
  const float* gam[4]  = {(const float*)d_in[17], (const float*)d_in[18],
                          (const float*)d_in[19], (const float*)d_in[20]};
  const float* bet[4]  = {(const float*)d_in[21], (const float*)d_in[22],
                          (const float*)d_in[23], (const float*)d_in[24]};
  const float* fcW[5]  = {(const float*)d_in[25], (const float*)d_in[26],
                          (const float*)d_in[27], (const float*)d_in[28],
                          (const float*)d_in[29]};
  const float* fcB[5]  = {(const float*)d_in[30], (const float*)d_in[31],
                          (const float*)d_in[32], (const float*)d_in[33],
                          (const float*)d_in[34]};
  const long long* ei    = (const long long*)d_in[35];
  const long long* batch = (const long long*)d_in[36];

  // workspace carve-out (256B aligned), ~111 MB total
  char* ws = (char*)d_ws;
  size_t off = 0;
  auto carve = [&](size_t bytes) -> char* {
    char* p = ws + off;
    off = (off + bytes + 255) & ~((size_t)255);
    return p;
  };
  float*    bufA   = (float*)carve((size_t)NN * DD * 4);
  float*    bufB   = (float*)carve((size_t)NN * DD * 4);
  float*    eval   = (float*)carve((size_t)ETOT * 4);
  float*    esrc   = (float*)carve((size_t)NN * 4);
  float*    edstv  = (float*)carve((size_t)NN * 4);
  unsigned* menc   = (unsigned*)carve((size_t)NN * 4);
  float*    ssum   = (float*)carve((size_t)NN * 4);
  float*    csum   = (float*)carve(DD * 4);
  float*    csq    = (float*)carve(DD * 4);
  float*    mu     = (float*)carve(DD * 4);
  float*    rstd   = (float*)carve(DD * 4);
  float*    pooled = (float*)carve((size_t)5 * GG * DD * 4);

  k_zero_f32<<<40, 256, 0, stream>>>(pooled, 5 * GG * DD);
  k_pool_x<<<STAT_BLOCKS, 128, 0, stream>>>(x, batch, pooled);  // JK head 0

  const float* hin = x;
  float* U = bufA;  // GEMM destination / final layer output
  float* V = bufB;  // aggregation (scatter) target
  for (int layer = 0; layer < NLAYER; layer++) {
    k_gemm128_wmma<<<NN / 32, 128, 0, stream>>>(hin, Ws[layer], U);
    k_row_scores<<<NN / 8, 256, 0, stream>>>(U, asrc[layer], adst[layer], esrc, edstv);
    k_init_ms<<<(NN + 255) / 256, 256, 0, stream>>>(menc, ssum);
    k_edge_max<<<(ETOT + 255) / 256, 256, 0, stream>>>(ei, esrc, edstv, eval, menc);
    k_edge_exp<<<(ETOT + 255) / 256, 256, 0, stream>>>(ei, eval, menc, ssum);
    k_zero_f32<<<2048, 256, 0, stream>>>(V, NN * DD);
    k_aggregate<<<ETOT / 2, 256, 0, stream>>>(ei, eval, ssum, U, V);
    k_zero2_128<<<1, 128, 0, stream>>>(csum, csq);
    k_bn_stats<<<STAT_BLOCKS, 128, 0, stream>>>(V, csum, csq);
    k_bn_finalize<<<1, 128, 0, stream>>>(csum, csq, mu, rstd);
    k_bn_apply_pool<<<STAT_BLOCKS, 128, 0, stream>>>(
        V, gam[layer], bet[layer], mu, rstd, U, batch,
        pooled + (size_t)(layer + 1) * GG * DD);
    // ping-pong: layer output lives in U; it becomes next input
    float* t = U;
    hin = t;
    U = V;
    V = t;
  }

  k_jk_out<<<GG, 32, 0, stream>>>(pooled, fcW[0], fcW[1], fcW[2], fcW[3],
                                  fcW[4], fcB[0], fcB[1], fcB[2], fcB[3],
                                  fcB[4], (float*)d_out);
}